// KPLoss_377957122199
// MI455X (gfx1250) — compile-verified
//
#include <hip/hip_runtime.h>
#include <hip/hip_bf16.h>

typedef float v2f __attribute__((ext_vector_type(2)));
typedef float v8f __attribute__((ext_vector_type(8)));

#define B_    8192
#define K_    400
#define NS_   20
#define SEC_  20
#define S_    (K_ / SEC_)      // 20 sections
#define NSEC  (B_ * S_)        // 163840 sections total
#define NGRP  (NSEC / 5)       // 32768 wave-groups of 5 sections (exact)
#define NROW  (B_ * K_)        // 3276800 CE rows

__device__ __forceinline__ float smooth_l1(float x) {
    float ax = fabsf(x);
    return ax < 1.0f ? 0.5f * x * x : ax - 0.5f;
}

__device__ __forceinline__ float wave_reduce(float v) {
    #pragma unroll
    for (int o = 16; o > 0; o >>= 1) v += __shfl_down(v, o, 32);
    return v;
}

__global__ void kp_init_kernel(float* ws) {
    if (threadIdx.x < 4) ws[threadIdx.x] = 0.0f;
}

// ---------------- cross-entropy over [NROW, 20] logits ----------------
__global__ __launch_bounds__(256) void kp_ce_kernel(const float* __restrict__ logits,
                                                    const int*   __restrict__ label,
                                                    float*       __restrict__ ws) {
    int row = blockIdx.x * blockDim.x + threadIdx.x;
    float nll = 0.0f;
    if (row < NROW) {
        const float4* p = (const float4*)(logits + (size_t)row * NS_);  // 80B rows, 16B aligned
        float x[NS_];
        #pragma unroll
        for (int i = 0; i < 5; ++i) {
            float4 t = p[i];
            x[4*i+0] = t.x; x[4*i+1] = t.y; x[4*i+2] = t.z; x[4*i+3] = t.w;
        }
        float m = x[0];
        #pragma unroll
        for (int i = 1; i < NS_; ++i) m = fmaxf(m, x[i]);
        float se = 0.0f;
        #pragma unroll
        for (int i = 0; i < NS_; ++i) se += __expf(x[i] - m);
        int lb = label[row];
        nll = __logf(se) + m - x[lb];
    }
    nll = wave_reduce(nll);
    if ((threadIdx.x & 31) == 0) atomicAdd(&ws[0], nll);
}

// ------- fused keypoint / center / Procrustes-rotation losses -------
// One wave per 5 consecutive sections; H matrices via v_wmma_f32_16x16x4_f32.
__global__ __launch_bounds__(256) void kp_proc_kernel(const float* __restrict__ P,
                                                      const float* __restrict__ G,
                                                      float*       __restrict__ ws) {
    __shared__ __align__(16) float sP[8][300];  // 5 sections * 20 pts * 3 (1200B rows, 16B aligned)
    __shared__ __align__(16) float sG[8][300];
    __shared__ float sPm[8][15];                // 5 sections * 3 coord means
    __shared__ float sGm[8][15];
    __shared__ float sD[8][256];                // 16x16 WMMA result dump
    __shared__ float sR[8][45];                 // 5 sections * 3x3 rotation

    const int wave = threadIdx.x >> 5;
    const int lane = threadIdx.x & 31;
    const int grp  = blockIdx.x * 8 + wave;     // [0, NGRP)
    const size_t base = (size_t)grp * 300;      // 5 sections are contiguous floats

    float kp_acc = 0.0f, cent_acc = 0.0f, rot_acc = 0.0f;

    // Stage tile into LDS as float4 (75 per matrix); fuse smooth-L1 keypoint loss.
    {
        const float4* P4 = (const float4*)(P + base);
        const float4* G4 = (const float4*)(G + base);
        float4* sP4 = (float4*)&sP[wave][0];
        float4* sG4 = (float4*)&sG[wave][0];
        for (int t = lane; t < 75; t += 32) {
            float4 p = P4[t];
            float4 g = G4[t];
            sP4[t] = p;
            sG4[t] = g;
            kp_acc += smooth_l1(p.x - g.x) + smooth_l1(p.y - g.y)
                    + smooth_l1(p.z - g.z) + smooth_l1(p.w - g.w);
        }
    }
    __syncthreads();

    // Means: 30 tasks = {P,G} x 5 sections x 3 coords.
    if (lane < 30) {
        int mtx = lane / 15;
        int idx = lane % 15;
        int s = idx / 3, i = idx % 3;
        const float* src = mtx ? &sG[wave][0] : &sP[wave][0];
        float sum = 0.0f;
        #pragma unroll
        for (int k = 0; k < SEC_; ++k) sum += src[s * 60 + k * 3 + i];
        float mean = sum * (1.0f / 20.0f);
        if (mtx) sGm[wave][idx] = mean; else sPm[wave][idx] = mean;
    }
    __syncthreads();

    // Center-alignment loss (means ready), then center the tiles in place
    // so the WMMA build and rot-loss pass read Pc/Gc directly.
    if (lane < 15) cent_acc = smooth_l1(sPm[wave][lane] - sGm[wave][lane]);
    for (int t = lane; t < 300; t += 32) {
        int mi = (t / 60) * 3 + (t % 3);        // section*3 + coord
        sP[wave][t] -= sPm[wave][mi];
        sG[wave][t] -= sGm[wave][mi];
    }
    __syncthreads();

    // ---- H_s = Gc_s^T * Pc_s for s=0..4 via WMMA f32 16x16x4 ----
    // A[m,k]: m = 3s+i -> Gc ; B[k,n]: n = 3s+j -> Pc. Row/col 15 unused:
    // clamp its index to 0 and multiply by 0 mask (branch-free, full EXEC).
    const int nrow  = lane & 15;
    const int khalf = lane >> 4;
    const bool valid = nrow < 15;
    const int   nn   = valid ? nrow : 0;        // v_cndmask, no branch
    const float msk  = valid ? 1.0f : 0.0f;
    const int colbase = (nn / 3) * 60 + (nn % 3);
    v8f acc = {0.f, 0.f, 0.f, 0.f, 0.f, 0.f, 0.f, 0.f};
    #pragma unroll
    for (int kb = 0; kb < SEC_; kb += 4) {
        v2f a, b;
        #pragma unroll
        for (int v = 0; v < 2; ++v) {
            int k = kb + 2 * khalf + v;         // A/B K-lane layout coincide
            a[v] = msk * sG[wave][colbase + k * 3];
            b[v] = msk * sP[wave][colbase + k * 3];
        }
        acc = __builtin_amdgcn_wmma_f32_16x16x4_f32(false, a, false, b,
                                                    (short)0, acc, false, false);
    }
    // D layout: VGPR v, lanes 0-15 -> M=v, N=lane; lanes 16-31 -> M=v+8, N=lane-16.
    #pragma unroll
    for (int v = 0; v < 8; ++v) sD[wave][(v + 8 * khalf) * 16 + nrow] = acc[v];
    __syncthreads();

    // ---- orthogonal polar factor R = U V^T via det-scaled Newton ----
    if (lane < 5) {
        int s = lane;
        float X[3][3];
        #pragma unroll
        for (int i = 0; i < 3; ++i)
            #pragma unroll
            for (int j = 0; j < 3; ++j)
                X[i][j] = sD[wave][(3 * s + i) * 16 + (3 * s + j)];
        #pragma unroll
        for (int it = 0; it < 10; ++it) {
            float c00 =  X[1][1]*X[2][2] - X[1][2]*X[2][1];
            float c01 = -(X[1][0]*X[2][2] - X[1][2]*X[2][0]);
            float c02 =  X[1][0]*X[2][1] - X[1][1]*X[2][0];
            float c10 = -(X[0][1]*X[2][2] - X[0][2]*X[2][1]);
            float c11 =  X[0][0]*X[2][2] - X[0][2]*X[2][0];
            float c12 = -(X[0][0]*X[2][1] - X[0][1]*X[2][0]);
            float c20 =  X[0][1]*X[1][2] - X[0][2]*X[1][1];
            float c21 = -(X[0][0]*X[1][2] - X[0][2]*X[1][0]);
            float c22 =  X[0][0]*X[1][1] - X[0][1]*X[1][0];
            float det = X[0][0]*c00 + X[0][1]*c01 + X[0][2]*c02;
            float ad  = fmaxf(fabsf(det), 1e-30f);
            float idet = (det < 0.0f ? -1.0f : 1.0f) / ad;
            float z    = 1.0f / cbrtf(ad);      // |det|^(-1/3) accel factor
            float zi   = 1.0f / z;
            float cof[3][3] = {{c00, c01, c02}, {c10, c11, c12}, {c20, c21, c22}};
            #pragma unroll
            for (int i = 0; i < 3; ++i)
                #pragma unroll
                for (int j = 0; j < 3; ++j)
                    X[i][j] = 0.5f * (z * X[i][j] + zi * cof[i][j] * idet); // X^-T = cof/det
        }
        #pragma unroll
        for (int i = 0; i < 3; ++i)
            #pragma unroll
            for (int j = 0; j < 3; ++j)
                sR[wave][s * 9 + i * 3 + j] = X[i][j];
    }
    __syncthreads();

    // Rotation loss: smooth_l1(Pc @ R - Gc) over 100 (s,k) pairs per wave.
    // Tiles already centered in LDS.
    for (int t = lane; t < 100; t += 32) {
        int s = t / SEC_, k = t % SEC_;
        float pc[3], gc[3];
        #pragma unroll
        for (int i = 0; i < 3; ++i) {
            pc[i] = sP[wave][s * 60 + k * 3 + i];
            gc[i] = sG[wave][s * 60 + k * 3 + i];
        }
        #pragma unroll
        for (int j = 0; j < 3; ++j) {
            float aj = pc[0] * sR[wave][s * 9 + 0 * 3 + j]
                     + pc[1] * sR[wave][s * 9 + 1 * 3 + j]
                     + pc[2] * sR[wave][s * 9 + 2 * 3 + j];
            rot_acc += smooth_l1(aj - gc[j]);
        }
    }

    kp_acc   = wave_reduce(kp_acc);
    rot_acc  = wave_reduce(rot_acc);
    cent_acc = wave_reduce(cent_acc);
    if (lane == 0) {
        atomicAdd(&ws[1], kp_acc);
        atomicAdd(&ws[2], rot_acc);
        atomicAdd(&ws[3], cent_acc);
    }
}

__global__ void kp_fin_kernel(const float* __restrict__ ws, float* __restrict__ out) {
    if (threadIdx.x == 0 && blockIdx.x == 0) {
        float section = ws[0] / (float)NROW;
        float keyp    = ws[1] / (float)(NROW * 3);
        float rot     = ws[2] / (float)(NROW * 3);
        float cent    = ws[3] / (float)(NSEC * 3);
        out[0] = 1.0f * section + 4.0f * keyp + 5.0f * rot + 6.0f * cent;
    }
}

extern "C" void kernel_launch(void* const* d_in, const int* in_sizes, int n_in,
                              void* d_out, int out_size, void* d_ws, size_t ws_size,
                              hipStream_t stream) {
    const float* pred_kp = (const float*)d_in[0];   // [B,K,3] f32
    const float* gt_kp   = (const float*)d_in[1];   // [B,K,3] f32
    const float* logits  = (const float*)d_in[2];   // [B,K,NS] f32
    const int*   labels  = (const int*)d_in[3];     // [B,K] int
    float* out = (float*)d_out;
    float* ws  = (float*)d_ws;

    kp_init_kernel<<<1, 32, 0, stream>>>(ws);
    kp_ce_kernel<<<NROW / 256, 256, 0, stream>>>(logits, labels, ws);
    kp_proc_kernel<<<NGRP / 8, 256, 0, stream>>>(pred_kp, gt_kp, ws);
    kp_fin_kernel<<<1, 32, 0, stream>>>(ws, out);
}